// GNN_2894807958000
// MI455X (gfx1250) — compile-verified
//
#include <hip/hip_runtime.h>
#include <math.h>
#include <stdint.h>

namespace {

constexpr int   kN     = 50000;
constexpr int   kE     = 800000;
constexpr int   kD     = 128;
constexpr float kSlope = 0.01f;
constexpr float kEps   = 1e-12f;

typedef __attribute__((ext_vector_type(2))) float v2f;
typedef __attribute__((ext_vector_type(8))) float v8f;

// ---------------- CSR build (edges are layer-invariant: built once per launch) ----------------

__global__ void zero_i32(int* __restrict__ p, int n) {
  int i = blockIdx.x * blockDim.x + threadIdx.x;
  if (i < n) p[i] = 0;
}

__global__ void degree_hist(const int* __restrict__ dst, int* __restrict__ deg) {
  int e = blockIdx.x * blockDim.x + threadIdx.x;
  if (e < kE) atomicAdd(&deg[dst[e]], 1);
}

// Single-block Hillis-Steele scan over kN bins -> exclusive rowptr (+ fill cursor copy).
__global__ __launch_bounds__(1024)
void scan_rowptr(const int* __restrict__ deg, int* __restrict__ rowptr, int* __restrict__ cur) {
  __shared__ int buf[1024];
  __shared__ int carry;
  const int tid = threadIdx.x;
  if (tid == 0) carry = 0;
  __syncthreads();
  for (int base = 0; base < kN; base += 1024) {
    int i = base + tid;
    int v = (i < kN) ? deg[i] : 0;
    buf[tid] = v;
    __syncthreads();
#pragma unroll
    for (int off = 1; off < 1024; off <<= 1) {
      int t = (tid >= off) ? buf[tid - off] : 0;
      __syncthreads();
      buf[tid] += t;
      __syncthreads();
    }
    int excl = carry + buf[tid] - v;
    if (i < kN) { rowptr[i] = excl; cur[i] = excl; }
    __syncthreads();
    if (tid == 0) carry += buf[1023];
    __syncthreads();
  }
  if (tid == 0) rowptr[kN] = carry;
}

__global__ void csr_fill(const int* __restrict__ src, const int* __restrict__ dst,
                         int* __restrict__ cur, int* __restrict__ eidx) {
  int e = blockIdx.x * blockDim.x + threadIdx.x;
  if (e < kE) {
    int pos = atomicAdd(&cur[dst[e]], 1);
    eidx[pos] = src[e];
  }
}

// ---------------- Gather-based segment mean: one wave32 per node, no float atomics ----------------

__global__ void aggregate_mean(const float* __restrict__ feat,
                               const int* __restrict__ eidx,
                               const int* __restrict__ rowptr,
                               float* __restrict__ mean) {
  int node = (int)(((long)blockIdx.x * blockDim.x + threadIdx.x) >> 5);
  int lane = threadIdx.x & 31;
  if (node >= kN) return;
  int beg = rowptr[node];
  int end = rowptr[node + 1];
  float4 acc = make_float4(0.f, 0.f, 0.f, 0.f);
  for (int e = beg; e < end; ++e) {
    int s = eidx[e];  // wave-uniform
    float4 v = reinterpret_cast<const float4*>(feat + (size_t)s * kD)[lane];
    acc.x += v.x; acc.y += v.y; acc.z += v.z; acc.w += v.w;
  }
  float rc = 1.0f / fmaxf((float)(end - beg), 1.0f);
  float4 o; o.x = acc.x * rc; o.y = acc.y * rc; o.z = acc.z * rc; o.w = acc.w * rc;
  reinterpret_cast<float4*>(mean + (size_t)node * kD)[lane] = o;
}

// ---------------- Fused dual-GEMM + bias + L2-norm + leaky/residual (WMMA f32 16x16x4) ----------------
// Block = 128 threads (4 waves) handles 16 rows; wave w owns col tiles 2w, 2w+1.

__global__ __launch_bounds__(128)
void sage_layer(const float* __restrict__ xin,
                const float* __restrict__ mean,
                const float* __restrict__ Wl,
                const float* __restrict__ bl,
                const float* __restrict__ Wr,
                const float* __restrict__ resid,
                float* __restrict__ outp,
                int apply_leaky) {
  __shared__ __align__(16) float s_mean[16][kD];
  __shared__ __align__(16) float s_x[16][kD];
  __shared__ float s_n2[16];

  const int tid  = threadIdx.x;
  const int row0 = blockIdx.x * 16;

  // Stage 16x128 mean and x tiles straight into LDS with async copies (ASYNCcnt path).
  {
    int r  = tid >> 3;          // 0..15
    int c0 = (tid & 7) * 16;    // 0,16,...,112
    const float* gm = mean + (size_t)(row0 + r) * kD + c0;
    const float* gx = xin  + (size_t)(row0 + r) * kD + c0;
    uint32_t lm = (uint32_t)(uintptr_t)&s_mean[r][c0];  // low 32 bits of flat shared ptr = LDS offset
    uint32_t lx = (uint32_t)(uintptr_t)&s_x[r][c0];
#pragma unroll
    for (int i = 0; i < 4; ++i) {
      uint64_t am = (uint64_t)(uintptr_t)(gm + i * 4);
      uint64_t ax = (uint64_t)(uintptr_t)(gx + i * 4);
      asm volatile("global_load_async_to_lds_b128 %0, %1, off"
                   :: "v"(lm + i * 16), "v"(am) : "memory");
      asm volatile("global_load_async_to_lds_b128 %0, %1, off"
                   :: "v"(lx + i * 16), "v"(ax) : "memory");
    }
    if (tid < 16) s_n2[tid] = 0.0f;
    asm volatile("s_wait_asynccnt 0" ::: "memory");
  }
  __syncthreads();

  const int wave = tid >> 5;
  const int lane = tid & 31;
  const int half = lane >> 4;   // K sub-pair selector per ISA f32 A/B layout
  const int l16  = lane & 15;   // M (for A) / N (for B)

  const int colA = (wave * 2 + 0) * 16;
  const int colB = (wave * 2 + 1) * 16;

  // B[k][n] = W[n][k]  (out = in @ W^T), W row-major [D][D].
  const float* wlA = Wl + (size_t)(colA + l16) * kD;
  const float* wrA = Wr + (size_t)(colA + l16) * kD;
  const float* wlB = Wl + (size_t)(colB + l16) * kD;
  const float* wrB = Wr + (size_t)(colB + l16) * kD;

  v8f acc0 = {};
  v8f acc1 = {};

#pragma unroll 4
  for (int k = 0; k < kD; k += 4) {
    const int kk = k + 2 * half;          // lanes 0-15: K=k,k+1; lanes 16-31: K=k+2,k+3
    v2f am, ax, b;
    am.x = s_mean[l16][kk]; am.y = s_mean[l16][kk + 1];
    ax.x = s_x[l16][kk];    ax.y = s_x[l16][kk + 1];

    b = *reinterpret_cast<const v2f*>(wlA + kk);
    acc0 = __builtin_amdgcn_wmma_f32_16x16x4_f32(false, am, false, b, (short)0, acc0, false, false);
    b = *reinterpret_cast<const v2f*>(wrA + kk);
    acc0 = __builtin_amdgcn_wmma_f32_16x16x4_f32(false, ax, false, b, (short)0, acc0, false, false);
    b = *reinterpret_cast<const v2f*>(wlB + kk);
    acc1 = __builtin_amdgcn_wmma_f32_16x16x4_f32(false, am, false, b, (short)0, acc1, false, false);
    b = *reinterpret_cast<const v2f*>(wrB + kk);
    acc1 = __builtin_amdgcn_wmma_f32_16x16x4_f32(false, ax, false, b, (short)0, acc1, false, false);
  }

  // Bias + per-row sum-of-squares (C layout: VGPR r -> row r (lanes 0-15) / r+8 (lanes 16-31)).
  const float biasA = bl[colA + l16];
  const float biasB = bl[colB + l16];
#pragma unroll
  for (int r = 0; r < 8; ++r) {
    acc0[r] += biasA;
    acc1[r] += biasB;
    float pa = acc0[r], pb = acc1[r];
    atomicAdd(&s_n2[r + 8 * half], pa * pa + pb * pb);
  }
  __syncthreads();

#pragma unroll
  for (int r = 0; r < 8; ++r) {
    const int row = row0 + r + 8 * half;
    const float inv = 1.0f / fmaxf(sqrtf(s_n2[r + 8 * half]), kEps);
    float va = acc0[r] * inv;
    float vb = acc1[r] * inv;
    if (apply_leaky) {
      va = va > 0.0f ? va : va * kSlope;
      vb = vb > 0.0f ? vb : vb * kSlope;
    }
    if (resid != nullptr) {
      va += resid[(size_t)row * kD + colA + l16];
      vb += resid[(size_t)row * kD + colB + l16];
    }
    outp[(size_t)row * kD + colA + l16] = va;
    outp[(size_t)row * kD + colB + l16] = vb;
  }
}

}  // namespace

extern "C" void kernel_launch(void* const* d_in, const int* in_sizes, int n_in,
                              void* d_out, int out_size, void* d_ws, size_t ws_size,
                              hipStream_t stream) {
  (void)in_sizes; (void)n_in; (void)out_size; (void)ws_size;

  const float* x   = (const float*)d_in[0];
  const int*   ei  = (const int*)d_in[1];   // [2, E]: src row then dst row
  const float* W1l = (const float*)d_in[2];
  const float* b1l = (const float*)d_in[3];
  const float* W1r = (const float*)d_in[4];
  const float* W2l = (const float*)d_in[5];
  const float* b2l = (const float*)d_in[6];
  const float* W2r = (const float*)d_in[7];
  float* out = (float*)d_out;

  const int* src = ei;
  const int* dst = ei + kE;

  // Workspace layout: mean[N*D] f32 | h1[N*D] f32 | rowptr[N+1] | deg[N] | cur[N] | eidx[E]  (~55 MB)
  float* meanbuf = (float*)d_ws;
  float* h1      = meanbuf + (size_t)kN * kD;
  int*   rowptr  = (int*)(h1 + (size_t)kN * kD);
  int*   deg     = rowptr + (kN + 1);
  int*   cur     = deg + kN;
  int*   eidx    = cur + kN;

  const int nb_n = (kN + 255) / 256;
  const int nb_e = (kE + 255) / 256;
  const int nb_w = (int)(((long)kN * 32 + 255) / 256);  // one wave per node
  const int nb_g = kN / 16;                             // 3125 row-blocks

  // ---- CSR build (once; edges shared by both layers) ----
  zero_i32   <<<nb_n, 256, 0, stream>>>(deg, kN);
  degree_hist<<<nb_e, 256, 0, stream>>>(dst, deg);
  scan_rowptr<<<1, 1024, 0, stream>>>(deg, rowptr, cur);
  csr_fill   <<<nb_e, 256, 0, stream>>>(src, dst, cur, eidx);

  // ---- Layer 1 ----
  aggregate_mean<<<nb_w, 256, 0, stream>>>(x, eidx, rowptr, meanbuf);
  sage_layer    <<<nb_g, 128, 0, stream>>>(x, meanbuf, W1l, b1l, W1r, nullptr, h1, 1);

  // ---- Layer 2 ----
  aggregate_mean<<<nb_w, 256, 0, stream>>>(h1, eidx, rowptr, meanbuf);
  sage_layer    <<<nb_g, 128, 0, stream>>>(h1, meanbuf, W2l, b2l, W2r, h1, out, 0);
}